// Semantic_Router_66571993088389
// MI455X (gfx1250) — compile-verified
//
#include <hip/hip_runtime.h>
#include <hip/hip_bf16.h>

typedef float v2f __attribute__((ext_vector_type(2)));
typedef float v8f __attribute__((ext_vector_type(8)));

#define B_   64
#define C_   384
#define NT   1024
#define NS   4096
#define HS   64        // 64x64 token grid
#define NWIN 1024      // 32x32 windows of 2x2
#define TOPK 256
#define EPSV 1e-12f

// ---------------- Kernel 1: per-batch channel mean + L2 normalize ----------------
__global__ void zbar_kernel(const float* __restrict__ z, float* __restrict__ zbar) {
    __shared__ float red[512];
    const int b = blockIdx.x;
    const int c = threadIdx.x;            // 0..383
    const float* zb = z + (size_t)b * NT * C_;
    float s = 0.f;
    for (int n = 0; n < NT; ++n) s += zb[(size_t)n * C_ + c];   // coalesced over c
    const float mean = s * (1.0f / (float)NT);
    red[c] = mean * mean;
    if (c < 128) red[384 + c] = 0.f;      // pad to 512
    __syncthreads();
    for (int st = 256; st > 0; st >>= 1) {
        if (c < st) red[c] += red[c + st];
        __syncthreads();
    }
    const float nrm = sqrtf(red[0]);
    zbar[b * C_ + c] = mean / fmaxf(nrm, EPSV);
}

// ---------------- Kernel 2: resp via V_WMMA_F32_16X16X4_F32 ----------------
// Block = 128 threads = 4 waves; each wave computes resp for 16 tokens.
// A (16x4 f32): lane L<16 -> row M=L, K=0(,1); lane L>=16 -> row M=L-16, K=2(,3).
// B (4x16 f32): zbar chunk broadcast across all 16 columns -> every D column
// holds the 16 dot products (column layout therefore irrelevant).
__global__ void resp_kernel(const float* __restrict__ x,
                            const float* __restrict__ zbar,
                            float* __restrict__ resp) {
    __shared__ float zl[C_];
    const int b    = blockIdx.x >> 6;     // 64 blocks per batch
    const int tile = blockIdx.x & 63;
    const int tid  = threadIdx.x;
    for (int c = tid; c < C_; c += 128) zl[c] = zbar[b * C_ + c];
    __syncthreads();

    const int wave = tid >> 5;
    const int lane = tid & 31;
    const int half = lane >> 4;           // 0: K=0,1  1: K=2,3
    const int m    = lane & 15;           // token row within tile
    const int t0   = tile * 64 + wave * 16;

    const float* row = x + ((size_t)b * NS + t0 + m) * C_ + 2 * half;

    v8f acc = {0.f, 0.f, 0.f, 0.f, 0.f, 0.f, 0.f, 0.f};
    float ss = 0.f;
    for (int k0 = 0; k0 < C_; k0 += 4) {
        v2f a;
        a.x = row[k0];
        a.y = row[k0 + 1];
        v2f bf;
        bf.x = zl[k0 + 2 * half];
        bf.y = zl[k0 + 2 * half + 1];
        ss += a.x * a.x + a.y * a.y;
        acc = __builtin_amdgcn_wmma_f32_16x16x4_f32(
            false, a, false, bf, (short)0, acc, false, false);
    }
    // full row self-dot: lane m has channels {0,1} mod 4, lane m+16 has {2,3}
    ss += __shfl_xor(ss, 16, 32);

    // extract dot for token m: D row i lives in acc[i] of lanes<16 (rows 0..7)
    // and acc[i] of lanes>=16 (rows 8..15); broadcast from lanes 0 and 16.
    float dot = 0.f;
#pragma unroll
    for (int i = 0; i < 8; ++i) {
        const float vlo = __shfl(acc[i], 0, 32);
        const float vhi = __shfl(acc[i], 16, 32);
        if (m == i)     dot = vlo;
        if (m == i + 8) dot = vhi;
    }
    if (lane < 16) {
        resp[(size_t)b * NS + t0 + lane] = dot / fmaxf(sqrtf(ss), EPSV);
    }
}

// ---------------- Kernel 3: 2x2 window mean of resp ----------------
__global__ void score_kernel(const float* __restrict__ resp, float* __restrict__ scores) {
    const int i = blockIdx.x * blockDim.x + threadIdx.x;
    if (i >= B_ * NWIN) return;
    const int b = i >> 10, win = i & 1023;
    const int wy = win >> 5, wx = win & 31;
    const float* rb = resp + (size_t)b * NS;
    const int n0 = (2 * wy) * HS + 2 * wx;
    scores[i] = 0.25f * (rb[n0] + rb[n0 + 1] + rb[n0 + HS] + rb[n0 + HS + 1]);
}

// ---------------- Kernel 4: per-batch bitonic sort (desc, stable ties) ----------------
__global__ void sort_kernel(const float* __restrict__ scores, int* __restrict__ order) {
    __shared__ float key[NWIN];
    __shared__ int   idx[NWIN];
    const int b = blockIdx.x, t = threadIdx.x;   // 512 threads
    for (int i = t; i < NWIN; i += 512) { key[i] = scores[b * NWIN + i]; idx[i] = i; }
    __syncthreads();
    for (int k = 2; k <= NWIN; k <<= 1) {
        for (int j = k >> 1; j > 0; j >>= 1) {
            const int i = ((t & ~(j - 1)) << 1) | (t & (j - 1));
            const int p = i | j;
            const bool desc = ((i & k) == 0);
            const float ka = key[i], kb = key[p];
            const int   ia = idx[i], ib = idx[p];
            // "a before b" in final order: higher score first; tie -> lower index
            const bool a_first = (ka > kb) || (ka == kb && ia < ib);
            const bool do_swap = desc ? (!a_first) : a_first;
            if (do_swap) { key[i] = kb; key[p] = ka; idx[i] = ib; idx[p] = ia; }
            __syncthreads();
        }
    }
    for (int i = t; i < NWIN; i += 512) order[b * NWIN + i] = idx[i];
}

// ---------------- Kernel 5: gather windows into (selected, non-selected) ----------------
// One block per (rank, batch); copies 4 token rows of 384 floats as float4.
__global__ void gather_kernel(const float* __restrict__ x,
                              const int* __restrict__ order,
                              float* __restrict__ out) {
    const int r = blockIdx.x;   // rank 0..1023
    const int b = blockIdx.y;   // batch
    const int t = threadIdx.x;  // 0..95 (96 float4 per row)
    const int win = order[b * NWIN + r];
    const int wy = win >> 5, wx = win & 31;

    const size_t SEL = (size_t)B_ * TOPK * 4 * C_;
    size_t dstoff;
    if (r < TOPK) dstoff = ((size_t)(b * TOPK + r)) * 4 * C_;
    else          dstoff = SEL + ((size_t)(b * (NWIN - TOPK) + (r - TOPK))) * 4 * C_;
    float4* dst4 = (float4*)(out + dstoff);

#pragma unroll
    for (int tt = 0; tt < 4; ++tt) {
        const int ty = tt >> 1, tx = tt & 1;
        const int n = (2 * wy + ty) * HS + (2 * wx + tx);
        const float4* s4 = (const float4*)(x + ((size_t)b * NS + n) * C_);
        dst4[tt * 96 + t] = s4[t];
    }
}

extern "C" void kernel_launch(void* const* d_in, const int* in_sizes, int n_in,
                              void* d_out, int out_size, void* d_ws, size_t ws_size,
                              hipStream_t stream) {
    const float* z = (const float*)d_in[0];   // (64,1024,384)
    const float* x = (const float*)d_in[1];   // (64,4096,384)
    float* out = (float*)d_out;

    float* zbar   = (float*)d_ws;                       // 64*384
    float* resp   = zbar + (size_t)B_ * C_;             // 64*4096
    float* scores = resp + (size_t)B_ * NS;             // 64*1024
    int*   order  = (int*)(scores + (size_t)B_ * NWIN); // 64*1024

    zbar_kernel  <<<B_, C_, 0, stream>>>(z, zbar);
    resp_kernel  <<<B_ * 64, 128, 0, stream>>>(x, zbar, resp);
    score_kernel <<<(B_ * NWIN + 255) / 256, 256, 0, stream>>>(resp, scores);
    sort_kernel  <<<B_, 512, 0, stream>>>(scores, order);
    gather_kernel<<<dim3(NWIN, B_), 96, 0, stream>>>(x, order, out);
}